// Representation_85452669321944
// MI455X (gfx1250) — compile-verified
//
#include <hip/hip_runtime.h>
#include <hip/hip_bf16.h>

typedef __attribute__((ext_vector_type(2))) float v2f;
typedef __attribute__((ext_vector_type(8))) float v8f;

// ---------------------------------------------------------------------------
// Generic fp32 GEMM using V_WMMA_F32_16X16X4_F32, N register blocking and
// explicit software pipelining (next K-step's fragments are loaded while the
// current step's WMMAs execute, so waits are partial instead of loadcnt==0).
// All memory accesses are integer-offset GEPs off the __restrict__ global
// kernel arguments (no pointer arrays!) so they lower to global_load_* and
// never demote to flat_load_* (which would couple LOADcnt with DScnt).
// C[M,N] = A[M,K] @ B[K,N] (+bias) (optional relu).
// One wave computes a 16 x (16*NB) output strip; the A fragment is reused
// across NB WMMAs per K-step.
// BT=true: B stored transposed, element (k,n) read from Bm[n*ldb + k].
// Guard is wave-uniform so EXEC is all-ones for every executing WMMA.
// ---------------------------------------------------------------------------
template<int NB, bool BT, bool RELU>
__global__ __launch_bounds__(128) void gemm_f32_wmma(
    const float* __restrict__ A, int lda,
    const float* __restrict__ Bm, int ldb,
    const float* __restrict__ bias,
    float* __restrict__ C, int ldc,
    int M, int N, int K)
{
    int tilesN = N / (16 * NB);
    int totalTiles = (M >> 4) * tilesN;
    int wave = blockIdx.x * (blockDim.x >> 5) + (threadIdx.x >> 5);
    if (wave >= totalTiles) return;              // wave-uniform exit
    int tm = wave / tilesN;
    int tn = wave - tm * tilesN;
    int lane = threadIdx.x & 31;
    int half = lane >> 4;                        // 0: lanes 0-15, 1: lanes 16-31
    int l16  = lane & 15;

    int bn[NB];
    long brow[NB];                               // per-nb fixed part of B offset
    #pragma unroll
    for (int nb = 0; nb < NB; ++nb) {
        bn[nb]   = (tn * NB + nb) * 16 + l16;
        brow[nb] = BT ? (long)bn[nb] * ldb : (long)bn[nb];
    }

    // Advancing integer offsets (strength-reduced induction variables).
    long ak = (long)(tm * 16 + l16) * lda + 2 * half;   // A element offset
    long bk = BT ? (long)(2 * half) : (long)(2 * half) * ldb;
    const long bkstep = BT ? 4 : 4 * (long)ldb;

    v8f acc[NB];
    #pragma unroll
    for (int nb = 0; nb < NB; ++nb)
        acc[nb] = (v8f){0.f, 0.f, 0.f, 0.f, 0.f, 0.f, 0.f, 0.f};

    // Prologue: load fragments for k0 = 0.
    v2f a_c = *(const v2f*)(A + ak);
    v2f b_c[NB];
    #pragma unroll
    for (int nb = 0; nb < NB; ++nb) {
        if (BT) {
            b_c[nb] = *(const v2f*)(Bm + brow[nb] + bk);
        } else {
            b_c[nb].x = Bm[bk + brow[nb]];
            b_c[nb].y = Bm[bk + ldb + brow[nb]];
        }
    }

    // Pipelined main loop (all K here are >= 8 and multiples of 4).
    for (int k0 = 0; k0 < K - 4; k0 += 4) {
        ak += 4;
        bk += bkstep;
        v2f a_n = *(const v2f*)(A + ak);
        v2f b_n[NB];
        #pragma unroll
        for (int nb = 0; nb < NB; ++nb) {
            if (BT) {
                b_n[nb] = *(const v2f*)(Bm + brow[nb] + bk);
            } else {
                b_n[nb].x = Bm[bk + brow[nb]];
                b_n[nb].y = Bm[bk + ldb + brow[nb]];
            }
        }
        #pragma unroll
        for (int nb = 0; nb < NB; ++nb)
            acc[nb] = __builtin_amdgcn_wmma_f32_16x16x4_f32(
                false, a_c, false, b_c[nb], (short)0, acc[nb], false, false);
        a_c = a_n;
        #pragma unroll
        for (int nb = 0; nb < NB; ++nb) b_c[nb] = b_n[nb];
    }
    // Epilogue K-step.
    #pragma unroll
    for (int nb = 0; nb < NB; ++nb)
        acc[nb] = __builtin_amdgcn_wmma_f32_16x16x4_f32(
            false, a_c, false, b_c[nb], (short)0, acc[nb], false, false);

    int m0 = tm * 16 + half * 8;                 // C: VGPR r -> rows r / r+8
    #pragma unroll
    for (int nb = 0; nb < NB; ++nb) {
        float bv = bias ? bias[bn[nb]] : 0.0f;
        #pragma unroll
        for (int r2 = 0; r2 < 8; ++r2) {
            float v = acc[nb][r2] + bv;
            if (RELU) v = fmaxf(v, 0.0f);
            C[(long)(m0 + r2) * ldc + bn[nb]] = v;
        }
    }
}

template<bool BT, bool RELU>
static inline void launch_gemm(const float* A, int lda, const float* Bm, int ldb,
                               const float* bias, float* C, int ldc,
                               int M, int N, int K, hipStream_t stream)
{
    if ((N & 63) == 0) {
        int tiles = (M / 16) * (N / 64);
        int blocks = (tiles + 3) / 4;            // 4 waves (128 thr) per block
        gemm_f32_wmma<4, BT, RELU><<<blocks, 128, 0, stream>>>(A, lda, Bm, ldb, bias, C, ldc, M, N, K);
    } else {                                      // N multiple of 32 (e.g. 288)
        int tiles = (M / 16) * (N / 32);
        int blocks = (tiles + 3) / 4;
        gemm_f32_wmma<2, BT, RELU><<<blocks, 128, 0, stream>>>(A, lda, Bm, ldb, bias, C, ldc, M, N, K);
    }
}

// ---------------------------------------------------------------------------
// r = flip(sinusoid(L=128, 32), 0) @ W_r   -> r[128,32]
// ---------------------------------------------------------------------------
__global__ __launch_bounds__(256) void rpos_kernel(const float* __restrict__ W_r,
                                                   float* __restrict__ r)
{
    int idx = blockIdx.x * 256 + threadIdx.x;    // 0..4095
    int p = idx >> 5, j = idx & 31;
    float pos = (float)(127 - p);                // flip on axis 0
    float acc = 0.f;
    for (int i = 0; i < 32; ++i) {
        float angle = pos * powf(10000.0f, -(float)i * (1.0f / 16.0f));
        float e = ((i & 1) == 0) ? sinf(angle) : cosf(angle);
        acc += e * W_r[i * 32 + j];
    }
    r[idx] = acc;
}

// X_last[b, 0:256] = hist_x[b, 127, :]; X_last[b, 256:288] = r[127, :]
__global__ __launch_bounds__(256) void xlast_kernel(const float* __restrict__ hist,
                                                    const float* __restrict__ r,
                                                    float* __restrict__ X)
{
    int idx = blockIdx.x * 256 + threadIdx.x;
    if (idx >= 512 * 288) return;
    int b = idx / 288, c = idx - b * 288;
    X[idx] = (c < 256) ? hist[(long)b * 128 * 256 + 127 * 256 + c]
                       : r[127 * 32 + (c - 256)];
}

// cc[b,h] = b_embed . T[h,b,:] + b_attn_k[h,:] . q[b,h,:]
__global__ __launch_bounds__(128) void cconst_kernel(const float* __restrict__ T,
                                                     const float* __restrict__ Q,
                                                     const float* __restrict__ b_embed,
                                                     const float* __restrict__ b_attn,
                                                     float* __restrict__ cc)
{
    __shared__ float red[128];
    int bh = blockIdx.x;
    int b = bh >> 2, h = bh & 3;
    int t = threadIdx.x;
    float s = 0.f;
    const float* Trow = T + ((long)h * 512 + b) * 512;
    for (int e = t; e < 512; e += 128) s += b_embed[e] * Trow[e];
    s += b_attn[512 + h * 128 + t] * Q[b * 512 + h * 128 + t];
    red[t] = s;
    __syncthreads();
    for (int off = 64; off > 0; off >>= 1) {
        if (t < off) red[t] += red[t + off];
        __syncthreads();
    }
    if (t == 0) cc[bh] = red[0];
}

// ---------------------------------------------------------------------------
// Per-batch attention (last query row only; mask provably inactive):
//   s[h,k] = X[b,k].u[b,h] + cc[b,h];  a = softmax(s/sqrt(128));
//   Z[b,h,:] = sum_k a[h,k] * X[b,k]   (X = [hist_x[b,k] | r[k]], 288 dims)
// ---------------------------------------------------------------------------
#define TS 257
__global__ __launch_bounds__(256) void attn_kernel(const float* __restrict__ hist,
                                                   const float* __restrict__ r,
                                                   const float* __restrict__ U,
                                                   const float* __restrict__ cc,
                                                   float* __restrict__ Z)
{
    __shared__ float u_s[4 * 288];
    __shared__ float tileX[16 * TS];
    __shared__ float s_s[4 * 128];
    __shared__ float a_s[4 * 128];
    __shared__ float mh[4], inv[4];

    int b = blockIdx.x;
    int t = threadIdx.x;

    for (int i = t; i < 4 * 288; i += 256) {
        int h = i / 288, c = i - h * 288;
        u_s[i] = U[(long)h * 512 * 288 + (long)b * 288 + c];
    }
    __syncthreads();

    if (t < 128) {
        int k = t;
        for (int h = 0; h < 4; ++h) {
            float acc = cc[b * 4 + h];
            for (int j = 0; j < 32; ++j) acc += r[k * 32 + j] * u_s[h * 288 + 256 + j];
            s_s[h * 128 + k] = acc;
        }
    }
    __syncthreads();

    const float* Xb = hist + (long)b * 128 * 256;

    // scores: hist part, 16 rows per LDS tile
    for (int k0 = 0; k0 < 128; k0 += 16) {
        for (int i = t; i < 16 * 256; i += 256) {
            int kk = i >> 8, c = i & 255;
            tileX[kk * TS + c] = Xb[(k0 + kk) * 256 + c];
        }
        __syncthreads();
        int part = t & 3;
        int out = t >> 2;                 // 16 rows x 4 heads = 64 outputs
        int kk = out & 15, h = out >> 4;
        float sum = 0.f;
        for (int i = 0; i < 64; ++i) {
            int c = part + 4 * i;
            sum += tileX[kk * TS + c] * u_s[h * 288 + c];
        }
        sum += __shfl_xor(sum, 1, 32);
        sum += __shfl_xor(sum, 2, 32);
        if (part == 0) s_s[h * 128 + k0 + kk] += sum;
        __syncthreads();
    }

    // softmax over k (no mask for the last query row)
    const float scale = 0.08838834764831845f;   // 1/sqrt(128)
    if (t < 4) {
        float m = -1e30f;
        for (int k = 0; k < 128; ++k) m = fmaxf(m, s_s[t * 128 + k]);
        mh[t] = m;
    }
    __syncthreads();
    for (int i = t; i < 512; i += 256) {
        int h = i >> 7;
        a_s[i] = expf((s_s[i] - mh[h]) * scale);
    }
    __syncthreads();
    if (t < 4) {
        float s = 0.f;
        for (int k = 0; k < 128; ++k) s += a_s[t * 128 + k];
        inv[t] = 1.0f / s;
    }
    __syncthreads();
    for (int i = t; i < 512; i += 256) a_s[i] *= inv[i >> 7];
    __syncthreads();

    // z accumulation over hist channels (thread t owns channel t)
    float accz[4] = {0.f, 0.f, 0.f, 0.f};
    for (int k0 = 0; k0 < 128; k0 += 16) {
        for (int i = t; i < 16 * 256; i += 256) {
            int kk = i >> 8, c = i & 255;
            tileX[kk * TS + c] = Xb[(k0 + kk) * 256 + c];
        }
        __syncthreads();
        for (int kk = 0; kk < 16; ++kk) {
            float xv = tileX[kk * TS + t];
            #pragma unroll
            for (int h = 0; h < 4; ++h) accz[h] += a_s[h * 128 + k0 + kk] * xv;
        }
        __syncthreads();
    }
    for (int h = 0; h < 4; ++h)
        Z[((long)(b * 4 + h)) * 288 + t] = accz[h];

    // r-channel part of z (channels 256..287)
    if (t < 32) {
        int j = t;
        for (int h = 0; h < 4; ++h) {
            float acc = 0.f;
            for (int k = 0; k < 128; ++k) acc += a_s[h * 128 + k] * r[k * 32 + j];
            Z[((long)(b * 4 + h)) * 288 + 256 + j] = acc;
        }
    }
}

// x = relu(x3 + skip); per-row min/max normalize to [-1,1]
__global__ __launch_bounds__(256) void finalize_kernel(const float* __restrict__ x3,
                                                       const float* __restrict__ sk,
                                                       float* __restrict__ out)
{
    __shared__ float rmin[256], rmax[256];
    int b = blockIdx.x, t = threadIdx.x;
    float v = fmaxf(x3[b * 256 + t] + sk[b * 256 + t], 0.0f);
    rmin[t] = v;
    rmax[t] = v;
    __syncthreads();
    for (int off = 128; off > 0; off >>= 1) {
        if (t < off) {
            rmin[t] = fminf(rmin[t], rmin[t + off]);
            rmax[t] = fmaxf(rmax[t], rmax[t + off]);
        }
        __syncthreads();
    }
    float mn = rmin[0], mx = rmax[0];
    out[b * 256 + t] = 2.0f * (v - mn) / (mx - mn) - 1.0f;
}

// ---------------------------------------------------------------------------
extern "C" void kernel_launch(void* const* d_in, const int* in_sizes, int n_in,
                              void* d_out, int out_size, void* d_ws, size_t ws_size,
                              hipStream_t stream)
{
    const float* hist    = (const float*)d_in[0];
    // d_in[1] = end_pos: provably unused (mask never hits the last query row)
    const float* W_embed = (const float*)d_in[2];
    const float* b_embed = (const float*)d_in[3];
    const float* W_r     = (const float*)d_in[4];
    const float* W_attn  = (const float*)d_in[5];
    const float* b_attn  = (const float*)d_in[6];
    const float* W_proj  = (const float*)d_in[7];
    const float* b_proj  = (const float*)d_in[8];
    const float* W_skip  = (const float*)d_in[9];
    const float* b_skip  = (const float*)d_in[10];
    const float* W1 = (const float*)d_in[11];
    const float* b1 = (const float*)d_in[12];
    const float* W2 = (const float*)d_in[13];
    const float* b2 = (const float*)d_in[14];
    const float* W5 = (const float*)d_in[15];
    const float* b5 = (const float*)d_in[16];

    float* ws = (float*)d_ws;
    float* r_    = ws; ws += 128 * 32;
    float* Xlast = ws; ws += 512 * 288;
    float* bigA  = ws; ws += 4 * 512 * 512;   // T_h, later reused as EZ
    float* E_    = ws; ws += 512 * 512;       // E_last, later reused as x1
    float* Q_    = ws; ws += 512 * 512;       // Q, later reused as x2
    float* ctx   = ws; ws += 512 * 512;
    float* outl  = ws; ws += 512 * 512;
    float* cc    = ws; ws += 512 * 4;
    float* U_    = ws; ws += 4 * 512 * 288;
    float* Z_    = ws; ws += 4 * 512 * 288;
    float* sskip = ws; ws += 512 * 256;
    float* x3    = ws; ws += 512 * 256;
    float* T_ = bigA;
    float* EZ = bigA;
    float* x1 = E_;
    float* x2 = Q_;

    rpos_kernel<<<16, 256, 0, stream>>>(W_r, r_);
    xlast_kernel<<<(512 * 288 + 255) / 256, 256, 0, stream>>>(hist, r_, Xlast);

    // E_last = X_last @ W_embed + b_embed          (512 x 512, K=288)
    launch_gemm<false, false>(Xlast, 288, W_embed, 512, b_embed, E_, 512, 512, 512, 288, stream);
    // Q = E_last @ W_attn[:, :512] + b_attn[:512]  (512 x 512, K=512)
    launch_gemm<false, false>(E_, 512, W_attn, 1536, b_attn, Q_, 512, 512, 512, 512, stream);
    // T_h = Q_h @ W_k_h^T                          (512 x 512, K=128), B transposed
    for (int h = 0; h < 4; ++h)
        launch_gemm<true, false>(Q_ + h * 128, 512, W_attn + 512 + h * 128, 1536,
                                 nullptr, T_ + h * 512 * 512, 512, 512, 512, 128, stream);
    cconst_kernel<<<2048, 128, 0, stream>>>(T_, Q_, b_embed, b_attn, cc);
    // U_h = T_h @ W_embed^T                        (512 x 288, K=512), B transposed
    for (int h = 0; h < 4; ++h)
        launch_gemm<true, false>(T_ + h * 512 * 512, 512, W_embed, 512,
                                 nullptr, U_ + h * 512 * 288, 288, 512, 288, 512, stream);

    // streaming attention over hist_x -> Z (B*4, 288)
    attn_kernel<<<512, 256, 0, stream>>>(hist, r_, U_, cc, Z_);

    // EZ = Z @ W_embed + b_embed                   (2048 x 512, K=288)
    launch_gemm<false, false>(Z_, 288, W_embed, 512, b_embed, EZ, 512, 2048, 512, 288, stream);
    // ctx[:, h*128:(h+1)*128] = EZ_h @ W_v_h + b_v_h
    for (int h = 0; h < 4; ++h)
        launch_gemm<false, false>(EZ + h * 512, 2048, W_attn + 1024 + h * 128, 1536,
                                  b_attn + 1024 + h * 128, ctx + h * 128, 512,
                                  512, 128, 512, stream);
    // out_last = ctx @ W_proj + b_proj
    launch_gemm<false, false>(ctx, 512, W_proj, 512, b_proj, outl, 512, 512, 512, 512, stream);
    // skip = hist_x[:,0,:] @ W_skip + b_skip  (A row stride = L*256)
    launch_gemm<false, false>(hist, 128 * 256, W_skip, 256, b_skip, sskip, 256, 512, 256, 256, stream);
    // MLP
    launch_gemm<false, true >(outl, 512, W1, 512, b1, x1, 512, 512, 512, 512, stream);
    launch_gemm<false, true >(x1, 512, W2, 512, b2, x2, 512, 512, 512, 512, stream);
    launch_gemm<false, false>(x2, 512, W5, 256, b5, x3, 256, 512, 256, 512, stream);

    finalize_kernel<<<512, 256, 0, stream>>>(x3, sskip, (float*)d_out);
}